// TransferFunction_31026843746638
// MI455X (gfx1250) — compile-verified
//
#include <hip/hip_runtime.h>
#include <hip/hip_bf16.h>
#include <math.h>

// ---------------------------------------------------------------------------
// Shapes (from reference):
//   x, gumbel     : (8, 128, 64)   f32   -> 1024 rows x 64
//   filter_bank   : (1, 128, 32768) f32
//   resonance     : (64, 128)      f32   (K=resolution, N=frames)
//   r = softmax(x+g) @ resonance   : (1024, 128) f32  -> lives in d_ws
//   out           : (8, 1, 32768)  f32
// ---------------------------------------------------------------------------

typedef __attribute__((ext_vector_type(2))) float v2f;
typedef __attribute__((ext_vector_type(8))) float v8f;

#define BATCH      8
#define BANDS      128
#define RES        64
#define NFRAMES    128
#define NSAMPLES   32768
#define NROWS      (BATCH * BANDS)   // 1024

// ---------------------------------------------------------------------------
// Kernel 1: fused row-softmax (64-wide) + WMMA GEMM (1024x64 @ 64x128).
// One block = one 16-row M-tile. 256 threads = 8 wave32s; wave w computes
// N-tile w (16 frames) with 16 x V_WMMA_F32_16X16X4_F32.
// ---------------------------------------------------------------------------
__global__ __launch_bounds__(256)
void tf_softmax_wmma_kernel(const float* __restrict__ x,
                            const float* __restrict__ gn,
                            const float* __restrict__ res,   // (64,128)
                            float* __restrict__ r_out)       // (1024,128)
{
    __shared__ float sG[16][65];     // softmax'd A tile, padded (bank spread)
    __shared__ float sB[RES][130];   // resonance, padded rows

    const int tid = threadIdx.x;
    const int mt  = blockIdx.x;              // M-tile: rows mt*16 .. mt*16+15

    // ---- stage resonance into LDS (32KB, cooperative) ----
    for (int idx = tid; idx < RES * NFRAMES; idx += 256) {
        sB[idx / NFRAMES][idx % NFRAMES] = res[idx];
    }

    // ---- softmax for 16 rows x 64 cols: 16 threads per row, 4 cols each ----
    {
        const int rt   = tid >> 4;           // row in tile 0..15
        const int l16  = tid & 15;           // lane within row group
        const int row  = mt * 16 + rt;       // global row (b*128+c)
        const int base = row * RES + l16 * 4;

        float v0 = x[base + 0] + gn[base + 0];
        float v1 = x[base + 1] + gn[base + 1];
        float v2 = x[base + 2] + gn[base + 2];
        float v3 = x[base + 3] + gn[base + 3];

        float mx = fmaxf(fmaxf(v0, v1), fmaxf(v2, v3));
        #pragma unroll
        for (int off = 1; off < 16; off <<= 1)
            mx = fmaxf(mx, __shfl_xor(mx, off, 32));   // stays in 16-lane half

        float e0 = __expf(v0 - mx);
        float e1 = __expf(v1 - mx);
        float e2 = __expf(v2 - mx);
        float e3 = __expf(v3 - mx);

        float sm = (e0 + e1) + (e2 + e3);
        #pragma unroll
        for (int off = 1; off < 16; off <<= 1)
            sm += __shfl_xor(sm, off, 32);

        const float inv = 1.0f / sm;
        sG[rt][l16 * 4 + 0] = e0 * inv;
        sG[rt][l16 * 4 + 1] = e1 * inv;
        sG[rt][l16 * 4 + 2] = e2 * inv;
        sG[rt][l16 * 4 + 3] = e3 * inv;
    }
    __syncthreads();

    // ---- WMMA: C(16x16) += A(16x4) * B(4x16), K = 64 in steps of 4 ----
    const int wv    = tid >> 5;        // wave id 0..7 -> N-tile
    const int lane  = tid & 31;
    const int half  = lane >> 4;       // 0: K=kb,kb+1   1: K=kb+2,kb+3
    const int ml    = lane & 15;       // A: row M       B: col N (within tile)
    const int ncol  = wv * 16 + ml;    // global frame column for B

    v8f acc = {};
    #pragma unroll
    for (int kb = 0; kb < RES; kb += 4) {
        v2f a, b;
        a.x = sG[ml][kb + 2 * half + 0];
        a.y = sG[ml][kb + 2 * half + 1];
        b.x = sB[kb + 2 * half + 0][ncol];
        b.y = sB[kb + 2 * half + 1][ncol];
        // D = A*B + C  (exact f32 path: V_WMMA_F32_16X16X4_F32)
        acc = __builtin_amdgcn_wmma_f32_16x16x4_f32(
            /*neg_a=*/false, a, /*neg_b=*/false, b,
            /*c_mod=*/(short)0, acc, /*reuse_a=*/false, /*reuse_b=*/false);
    }

    // ---- store D: lane 0-15 -> M=v, lane 16-31 -> M=v+8 ----
    #pragma unroll
    for (int v = 0; v < 8; ++v) {
        const int mrow = v + 8 * half;
        r_out[(mt * 16 + mrow) * NFRAMES + ncol] = acc[v];
    }
}

// ---------------------------------------------------------------------------
// Kernel 2: linear-interp upsample (x256) + band-weighted mean.
// Block = 256 consecutive samples for one batch. Only frames
// {blk-1, blk, blk+1} are needed -> 3x128 floats in LDS, broadcast reads.
// filter_bank loads are stride-1 coalesced; fb (16.8MB) lives in L2 across
// the 8 batch passes.
// ---------------------------------------------------------------------------
__global__ __launch_bounds__(256)
void tf_interp_reduce_kernel(const float* __restrict__ r,   // (1024,128) = (b,c,f)
                             const float* __restrict__ fb,  // (128,32768)
                             float* __restrict__ out)       // (8,32768)
{
    __shared__ float sR[3 * BANDS];

    const int tid = threadIdx.x;
    const int blk = blockIdx.x;          // sample block 0..127
    const int b   = blockIdx.y;          // batch 0..7

    // stage frames blk-1, blk, blk+1 (clamped) for all 128 channels
    for (int idx = tid; idx < 3 * BANDS; idx += 256) {
        const int fslot = idx / BANDS;                  // 0..2
        const int c     = idx % BANDS;
        int f = blk - 1 + fslot;
        f = max(0, min(f, NFRAMES - 1));
        sR[fslot * BANDS + c] = r[(b * BANDS + c) * NFRAMES + f];
    }
    __syncthreads();

    const int i = blk * 256 + tid;       // sample index
    // pos = (i+0.5)/256 - 0.5, clipped to [0, 127]
    float pos = ((float)i + 0.5f) * (1.0f / 256.0f) - 0.5f;
    pos = fminf(fmaxf(pos, 0.0f), (float)(NFRAMES - 1));
    const int   lo = (int)floorf(pos);
    const int   hi = min(lo + 1, NFRAMES - 1);
    const float w  = pos - (float)lo;
    const float wl = 1.0f - w;
    const int slo = (lo - (blk - 1)) * BANDS;   // LDS slot base for lo
    const int shi = (hi - (blk - 1)) * BANDS;   // LDS slot base for hi

    float acc = 0.0f;
    #pragma unroll 8
    for (int c = 0; c < BANDS; ++c) {
        const float up = sR[slo + c] * wl + sR[shi + c] * w;
        acc += up * fb[c * NSAMPLES + i];
    }

    out[b * NSAMPLES + i] = acc * (1.0f / (float)BANDS);
}

// ---------------------------------------------------------------------------
extern "C" void kernel_launch(void* const* d_in, const int* in_sizes, int n_in,
                              void* d_out, int out_size, void* d_ws, size_t ws_size,
                              hipStream_t stream) {
    const float* x   = (const float*)d_in[0];   // (8,128,64)
    const float* gn  = (const float*)d_in[1];   // (8,128,64)
    const float* fb  = (const float*)d_in[2];   // (1,128,32768)
    const float* res = (const float*)d_in[3];   // (64,128)
    float* out  = (float*)d_out;                // (8,1,32768)
    float* r_ws = (float*)d_ws;                 // (1024,128) = 512KB scratch

    // Kernel 1: 64 M-tiles of 16 rows, 8 waves (N-tiles) each.
    tf_softmax_wmma_kernel<<<NROWS / 16, 256, 0, stream>>>(x, gn, res, r_ws);

    // Kernel 2: 128 sample-blocks x 8 batches.
    tf_interp_reduce_kernel<<<dim3(NSAMPLES / 256, BATCH), 256, 0, stream>>>(r_ws, fb, out);
}